// CameraLoss_25477746000340
// MI455X (gfx1250) — compile-verified
//
#include <hip/hip_runtime.h>
#include <math.h>

#define Bb 16
#define Hh 288
#define Ww 384
#define Nn (Hh*Ww)                 // 110592
#define BLK_PER_BATCH 16
#define WAVES_PER_BLK 8
#define THREADS 256
#define WAVES_PER_BATCH (BLK_PER_BATCH*WAVES_PER_BLK)   // 128
#define CHUNKS_PER_WAVE (Nn/32/WAVES_PER_BATCH)         // 27
#define NUM_ITER 5
#define LM_DAMP 1e-4f
#define Z_MIN 0.01f

typedef float v2f __attribute__((ext_vector_type(2)));
typedef float v8f __attribute__((ext_vector_type(8)));

// workspace layout (floats)
#define OFF_R 0                    // 16*9
#define OFF_T (Bb*9)               // 16*3
#define OFF_PART (Bb*9 + Bb*3)     // 16*16*49 block partials
#define PART_STRIDE 49

__global__ void init_state(float* ws) {
    int t = threadIdx.x;
    if (t < Bb) {
        float* R = ws + OFF_R + t*9;
        R[0]=1.f; R[1]=0.f; R[2]=0.f;
        R[3]=0.f; R[4]=1.f; R[5]=0.f;
        R[6]=0.f; R[7]=0.f; R[8]=1.f;
        float* tt = ws + OFF_T + t*3;
        tt[0]=0.f; tt[1]=0.f; tt[2]=0.f;
    }
}

__global__ __launch_bounds__(THREADS)
void accum_kernel(const float* __restrict__ pts3d, const float* __restrict__ conf,
                  const float* __restrict__ Kin, float* __restrict__ ws)
{
    // Wave-private staging: 64 rows x 8 floats per wave (2KB/wave)
    __shared__ float stage[WAVES_PER_BLK][64*8];
    __shared__ float red[WAVES_PER_BLK][49];

    const int batch = blockIdx.x / BLK_PER_BATCH;
    const int blk   = blockIdx.x % BLK_PER_BATCH;
    const int wave  = threadIdx.x >> 5;
    const int lane  = threadIdx.x & 31;
    const int wib   = blk*WAVES_PER_BLK + wave;   // 0..127 within batch

    const float* Kb = Kin + batch*9;
    const float fx = Kb[0], fy = Kb[4], cx = Kb[2], cy = Kb[5];
    const float* Rw = ws + OFF_R + batch*9;
    const float R00=Rw[0],R01=Rw[1],R02=Rw[2],
                R10=Rw[3],R11=Rw[4],R12=Rw[5],
                R20=Rw[6],R21=Rw[7],R22=Rw[8];
    const float* tw = ws + OFF_T + batch*3;
    const float t0=tw[0], t1=tw[1], t2=tw[2];
    // delta = 0.1*sqrt(var(gx)+var(gy)); var of uniform ints 0..n-1 = (n^2-1)/12
    const float DELTA = 0.1f * sqrtf(((float)Ww*(float)Ww - 1.0f)/12.0f +
                                     ((float)Hh*(float)Hh - 1.0f)/12.0f);

    float* myst = stage[wave];
    const float* P = pts3d + (size_t)batch*Nn*3;
    const float* C = conf  + (size_t)batch*Nn;

    v8f acc = {0.f,0.f,0.f,0.f,0.f,0.f,0.f,0.f};
    const int h = lane >> 4, f = lane & 15;

    for (int cc = 0; cc < CHUNKS_PER_WAVE; ++cc) {
        const int n = (wib + cc*WAVES_PER_BATCH)*32 + lane;
        const float Xx = P[n*3+0], Xy = P[n*3+1], Xz = P[n*3+2];
        const float cf = C[n];
        const float px = (float)(n % Ww), py = (float)(n / Ww);

        const float Xc = R00*Xx + R01*Xy + R02*Xz + t0;
        const float Yc = R10*Xx + R11*Xy + R12*Xz + t1;
        const float Zc = R20*Xx + R21*Xy + R22*Xz + t2;
        const float zc = fmaxf(Zc, Z_MIN);
        const float iz = 1.0f / zc;
        const float r0 = fx*Xc*iz + cx - px;
        const float r1 = fy*Yc*iz + cy - py;
        const float rn  = cf * sqrtf(r0*r0 + r1*r1);
        const float rob = sqrtf(fminf(1.0f, DELTA / fmaxf(rn, 1e-8f)));
        const float wt  = cf * rob;
        const float a  = fx*iz, bq = -fx*Xc*iz*iz;
        const float c2 = fy*iz, d2 = -fy*Yc*iz*iz;

        // two augmented rows of S = [Jw | wt*r], padded to 8 floats
        float4 r0a = make_float4(wt*a, 0.0f, wt*bq, wt*(bq*Yc));
        float4 r0b = make_float4(wt*(a*Zc - bq*Xc), wt*(-a*Yc), wt*r0, 0.0f);
        float4 r1a = make_float4(0.0f, wt*c2, wt*d2, wt*(d2*Yc - c2*Zc));
        float4 r1b = make_float4(wt*(-d2*Xc), wt*(c2*Xc), wt*r1, 0.0f);
        float4* sp = (float4*)(myst + lane*16);
        sp[0]=r0a; sp[1]=r0b; sp[2]=r1a; sp[3]=r1b;

        // LDS ops from one wave are processed in order; make sure the
        // compiler doesn't reorder and the data is ready before the reads.
        asm volatile("s_wait_dscnt 0x0" ::: "memory");

        // 16 K=4 chunks of S^T*S; A and B register images are identical
        // for this symmetric product (ISA 16x4 A / 4x16 B 32-bit layouts).
        #pragma unroll
        for (int j = 0; j < 16; ++j) {
            v2f av;
            av.x = myst[(4*j + 0 + 2*h)*8 + f];
            av.y = myst[(4*j + 1 + 2*h)*8 + f];
            acc = __builtin_amdgcn_wmma_f32_16x16x4_f32(
                      false, av, false, av, (short)0, acc, false, false);
        }
    }

    // D[m][n], m,n in 0..6: lanes 0..6 (N=lane), vgpr m (M=m)
    if (lane < 7) {
        #pragma unroll
        for (int m = 0; m < 7; ++m)
            red[wave][m*7 + lane] = acc[m];
    }
    __syncthreads();

    float* part = ws + OFF_PART + (size_t)(batch*BLK_PER_BATCH + blk)*PART_STRIDE;
    for (int t = threadIdx.x; t < 49; t += THREADS) {
        float s = 0.f;
        #pragma unroll
        for (int w = 0; w < WAVES_PER_BLK; ++w) s += red[w][t];
        part[t] = s;   // plain store: deterministic, per-block slot
    }
}

__global__ void solve_update(float* ws) {
    const int b = threadIdx.x;
    if (b >= Bb) return;
    float s49[49];
    const float* p = ws + OFF_PART + (size_t)b*BLK_PER_BATCH*PART_STRIDE;
    #pragma unroll
    for (int i = 0; i < 49; ++i) {
        float s = 0.f;
        for (int k = 0; k < BLK_PER_BATCH; ++k) s += p[k*PART_STRIDE + i];
        s49[i] = s;
    }
    float M[6][6], g[6];
    #pragma unroll
    for (int i = 0; i < 6; ++i) {
        #pragma unroll
        for (int j = 0; j < 6; ++j) M[i][j] = s49[i*7+j];
        g[i] = s49[i*7+6];
    }
    const float tr = M[0][0]+M[1][1]+M[2][2]+M[3][3]+M[4][4]+M[5][5];
    const float damp = LM_DAMP*tr/6.0f + 1e-6f;
    #pragma unroll
    for (int i = 0; i < 6; ++i) M[i][i] += damp;

    float x[6];
    #pragma unroll
    for (int i = 0; i < 6; ++i) x[i] = -g[i];
    // Gaussian elimination (SPD + damping: no pivoting needed)
    for (int i = 0; i < 6; ++i) {
        const float ip = 1.0f / M[i][i];
        for (int j = i+1; j < 6; ++j) {
            const float fct = M[j][i]*ip;
            for (int k = i; k < 6; ++k) M[j][k] -= fct*M[i][k];
            x[j] -= fct*x[i];
        }
    }
    for (int i = 5; i >= 0; --i) {
        float s = x[i];
        for (int j = i+1; j < 6; ++j) s -= M[i][j]*x[j];
        x[i] = s / M[i][i];
    }

    float* tp = ws + OFF_T + b*3;
    tp[0] += x[0]; tp[1] += x[1]; tp[2] += x[2];

    // R = exp_so3(dx[3:]) @ R
    const float wx=x[3], wy=x[4], wz=x[5];
    float th = sqrtf(wx*wx + wy*wy + wz*wz);
    th = fmaxf(th, 1e-8f);
    const float kx=wx/th, ky=wy/th, kz=wz/th;
    const float st=sinf(th), vc=1.0f-cosf(th);
    const float E00 = 1.0f - vc*(ky*ky+kz*kz);
    const float E01 = -st*kz + vc*kx*ky;
    const float E02 =  st*ky + vc*kx*kz;
    const float E10 =  st*kz + vc*kx*ky;
    const float E11 = 1.0f - vc*(kx*kx+kz*kz);
    const float E12 = -st*kx + vc*ky*kz;
    const float E20 = -st*ky + vc*kx*kz;
    const float E21 =  st*kx + vc*ky*kz;
    const float E22 = 1.0f - vc*(kx*kx+ky*ky);

    float* R = ws + OFF_R + b*9;
    const float r00=R[0],r01=R[1],r02=R[2],
                r10=R[3],r11=R[4],r12=R[5],
                r20=R[6],r21=R[7],r22=R[8];
    R[0]=E00*r00+E01*r10+E02*r20; R[1]=E00*r01+E01*r11+E02*r21; R[2]=E00*r02+E01*r12+E02*r22;
    R[3]=E10*r00+E11*r10+E12*r20; R[4]=E10*r01+E11*r11+E12*r21; R[5]=E10*r02+E11*r12+E12*r22;
    R[6]=E20*r00+E21*r10+E22*r20; R[7]=E20*r01+E21*r11+E22*r21; R[8]=E20*r02+E21*r12+E22*r22;
}

__global__ void loss_kernel(const float* __restrict__ pose,
                            const float* __restrict__ ws, float* out)
{
    __shared__ float rotl[Bb], trl[Bb];
    const int b = threadIdx.x;
    if (b < Bb) {
        // ---- gt_rel = inv(pose0) @ pose_b
        const float* P0 = pose;
        const float* Pb = pose + b*16;
        float Rg[3][3], tg[3];
        for (int i = 0; i < 3; ++i) {
            for (int j = 0; j < 3; ++j) {
                float s = 0.f;
                for (int k = 0; k < 3; ++k) s += P0[k*4+i]*Pb[k*4+j];   // R0^T Rb
                Rg[i][j] = s;
            }
            float s = 0.f;
            for (int k = 0; k < 3; ++k) s += P0[k*4+i]*(Pb[k*4+3] - P0[k*4+3]);
            tg[i] = s;
        }
        // ---- pred_rel = inv(T0) @ Tb from (R,t) state
        const float* R0 = ws + OFF_R;           // batch 0
        const float* Rb = ws + OFF_R + b*9;
        const float* t0 = ws + OFF_T;
        const float* tb = ws + OFF_T + b*3;
        float Rp[3][3], tp[3];
        for (int i = 0; i < 3; ++i) {
            for (int j = 0; j < 3; ++j) {
                float s = 0.f;
                for (int k = 0; k < 3; ++k) s += R0[k*3+i]*Rb[k*3+j];
                Rp[i][j] = s;
            }
            float s = 0.f;
            for (int k = 0; k < 3; ++k) s += R0[k*3+i]*(tb[k] - t0[k]);
            tp[i] = s;
        }
        // trace(Rp^T Rg) = sum elementwise Rp.*Rg
        float trace = 0.f;
        for (int i = 0; i < 3; ++i)
            for (int j = 0; j < 3; ++j) trace += Rp[i][j]*Rg[i][j];
        float cosang = 0.5f*(trace - 1.0f);
        cosang = fminf(fmaxf(cosang, -1.0f + 1e-7f), 1.0f - 1e-7f);
        rotl[b] = acosf(cosang);
        const float dx = tp[0]-tg[0], dy = tp[1]-tg[1], dz = tp[2]-tg[2];
        trl[b] = sqrtf(dx*dx + dy*dy + dz*dz);
    }
    __syncthreads();
    if (threadIdx.x == 0) {
        float sr = 0.f, st2 = 0.f;
        for (int i = 0; i < Bb; ++i) { sr += rotl[i]; st2 += trl[i]; }
        sr /= (float)Bb; st2 /= (float)Bb;
        out[0] = sr + st2;
        out[1] = sr;
        out[2] = st2;
    }
}

extern "C" void kernel_launch(void* const* d_in, const int* in_sizes, int n_in,
                              void* d_out, int out_size, void* d_ws, size_t ws_size,
                              hipStream_t stream) {
    const float* pts3d = (const float*)d_in[0];
    const float* conf  = (const float*)d_in[1];
    const float* Kin   = (const float*)d_in[2];
    const float* pose  = (const float*)d_in[3];
    float* ws  = (float*)d_ws;
    float* out = (float*)d_out;

    init_state<<<1, 32, 0, stream>>>(ws);
    for (int it = 0; it < NUM_ITER; ++it) {
        accum_kernel<<<Bb*BLK_PER_BATCH, THREADS, 0, stream>>>(pts3d, conf, Kin, ws);
        solve_update<<<1, 16, 0, stream>>>(ws);
    }
    loss_kernel<<<1, 16, 0, stream>>>(pose, ws, out);
}